// SeqAugment_17892833755543
// MI455X (gfx1250) — compile-verified
//
#include <hip/hip_runtime.h>
#include <stdint.h>

#define NB 128
#define NL 2048
#define ND 128

// ---------------------------------------------------------------------------
// Threefry-2x32 (exact JAX schedule): 20 rounds, parity const 0x1BD11BDA.
// ---------------------------------------------------------------------------
__device__ __forceinline__ void tf2x32(uint32_t k0, uint32_t k1,
                                       uint32_t x0, uint32_t x1,
                                       uint32_t& o0, uint32_t& o1) {
  uint32_t ks0 = k0, ks1 = k1, ks2 = k0 ^ k1 ^ 0x1BD11BDAu;
  x0 += ks0; x1 += ks1;
#define TFR(r) { x0 += x1; x1 = (x1 << (r)) | (x1 >> (32 - (r))); x1 ^= x0; }
  TFR(13) TFR(15) TFR(26) TFR(6)   x0 += ks1; x1 += ks2 + 1u;
  TFR(17) TFR(29) TFR(16) TFR(24)  x0 += ks2; x1 += ks0 + 2u;
  TFR(13) TFR(15) TFR(26) TFR(6)   x0 += ks0; x1 += ks1 + 3u;
  TFR(17) TFR(29) TFR(16) TFR(24)  x0 += ks1; x1 += ks2 + 4u;
  TFR(13) TFR(15) TFR(26) TFR(6)   x0 += ks2; x1 += ks0 + 5u;
#undef TFR
  o0 = x0; o1 = x1;
}

// jax.random.randint(key, (), 0, span): draws 64 bits = threefry(key,(0,1)),
// offset = ((hi % span) * (2^32 % span) + lo % span) % span  (uint32 wrap).
__device__ __forceinline__ uint32_t jax_randint(uint32_t k0, uint32_t k1,
                                                uint32_t span) {
  uint32_t hi, lo;
  tf2x32(k0, k1, 0u, 1u, hi, lo);
  uint32_t m = 65536u % span;
  m = (m * m) % span;                 // == 2^32 % span
  uint32_t off = (hi % span) * m + (lo % span);
  return off % span;
}

// jax.random.uniform bit pattern: ((bits>>9) | 0x3f800000) as float, minus 1.
__device__ __forceinline__ float bits_to_unit(uint32_t b) {
  return __uint_as_float((b >> 9) | 0x3f800000u) - 1.0f;
}

// ---------------------------------------------------------------------------
// Kernel 1: per-example plan. Emits rowsrc[b][i]: >=0 gather index, -1 zero
// row (crop tail), -2 mask row. Also writes aug_len (as float) to out tail.
// 1024 threads so the O(length^2) stable-rank count for the mask branch is
// ~2K LDS-broadcast iterations per thread.
// ---------------------------------------------------------------------------
__global__ __launch_bounds__(1024)
void aug_plan_kernel(const int* __restrict__ seq_len,
                     int* __restrict__ rowsrc,
                     float* __restrict__ out_len) {
  __shared__ float    sh_u[NL];
  __shared__ int      sh_i[8];
  __shared__ uint32_t sh_k[10];

  const int b = blockIdx.x;
  const int t = threadIdx.x;

  if (t == 0) {
    // split(key(42), 128): counts = iota(256) -> pairs (i, i+128); out
    // concat(x0',x1') reshaped (128,2) -> key_b = (out[2b], out[2b+1]).
    uint32_t kb0, kb1, d0, d1, e0, e1;
    uint32_t c0 = 2u * (uint32_t)b;
    if (b < 64) {
      tf2x32(0u, 42u, c0,      c0 + 128u, d0, d1);
      tf2x32(0u, 42u, c0 + 1u, c0 + 129u, e0, e1);
      kb0 = d0; kb1 = e0;               // both first-halves
    } else {
      uint32_t j = c0 - 128u;
      tf2x32(0u, 42u, j,      j + 128u, d0, d1);
      tf2x32(0u, 42u, j + 1u, j + 129u, e0, e1);
      kb0 = d1; kb1 = e1;               // both second-halves
    }
    // split(key_b, 5): counts iota(10) -> pairs (q, q+5), q=0..4.
    uint32_t Ex[5], Ey[5];
    for (int q = 0; q < 5; ++q)
      tf2x32(kb0, kb1, (uint32_t)q, (uint32_t)q + 5u, Ex[q], Ey[q]);
    uint32_t K[10] = { Ex[0], Ex[1],   // k1
                       Ex[2], Ex[3],   // k2
                       Ex[4], Ey[0],   // k3
                       Ey[1], Ey[2],   // k4
                       Ey[3], Ey[4] }; // k5
    for (int q = 0; q < 10; ++q) sh_k[q] = K[q];

    int length   = seq_len[b];
    float lf     = (float)length;
    int num_left = (int)floorf(lf * 0.2f);
    int num_mask = (int)floorf(lf * 0.6f);
    int num_re   = (int)floorf(lf * 0.2f);
    int crop_begin = (int)jax_randint(K[0], K[1],
                                      (uint32_t)max(length - num_left, 1));
    int re_begin   = (int)jax_randint(K[4], K[5],
                                      (uint32_t)max(length - num_re, 1));
    int method     = (int)jax_randint(K[8], K[9], 3u);
    sh_i[0] = method;   sh_i[1] = num_left; sh_i[2] = crop_begin;
    sh_i[3] = num_mask; sh_i[4] = re_begin; sh_i[5] = num_re;
    sh_i[6] = length;
    out_len[b] = (float)((method == 0) ? num_left : length);
  }
  __syncthreads();

  const int method = sh_i[0], num_left = sh_i[1], crop_begin = sh_i[2];
  const int num_mask = sh_i[3], re_begin = sh_i[4], num_re = sh_i[5];
  const int length = sh_i[6];
  int* rs = rowsrc + (size_t)b * NL;
  const float INF = __int_as_float(0x7f800000);

  if (method == 0) {
    // crop: rows [0,num_left) gather clip(crop_begin+i, 0, NL-1); rest zero.
    for (int i = t; i < NL; i += 1024)
      rs[i] = (i < num_left) ? min(crop_begin + i, NL - 1) : -1;
    return;
  }

  if (method == 1) {
    // mask: u = uniform(k2, (NL,)); invalid -> inf; stable rank < num_mask.
    uint32_t bx, by;
    tf2x32(sh_k[2], sh_k[3], (uint32_t)t, (uint32_t)t + 1024u, bx, by);
    sh_u[t]        = (t < length)        ? bits_to_unit(bx) : INF;
    sh_u[t + 1024] = (t + 1024 < length) ? bits_to_unit(by) : INF;
    __syncthreads();
    float u0 = sh_u[t], u1 = sh_u[t + 1024];
    int c0 = 0, c1 = 0;
    for (int j = 0; j < length; ++j) {        // LDS broadcast across threads
      float uj = sh_u[j];
      c0 += (int)((uj < u0) | ((uj == u0) & (j < t)));
      c1 += (int)((uj < u1) | ((uj == u1) & (j < t + 1024)));
    }
    rs[t]        = (c0 < num_mask) ? -2 : t;
    rs[t + 1024] = (c1 < num_mask) ? -2 : (t + 1024);
    return;
  }

  // reorder: u2 = uniform(k4) inside window, inf outside; in-window row i
  // gathers from re_begin + stable_rank(u2[i]).
  {
    uint32_t bx, by;
    tf2x32(sh_k[6], sh_k[7], (uint32_t)t, (uint32_t)t + 1024u, bx, by);
    const int i0 = t, i1 = t + 1024, end = re_begin + num_re;
    bool w0 = (i0 >= re_begin) & (i0 < end);
    bool w1 = (i1 >= re_begin) & (i1 < end);
    sh_u[i0] = w0 ? bits_to_unit(bx) : INF;
    sh_u[i1] = w1 ? bits_to_unit(by) : INF;
    __syncthreads();
#pragma unroll
    for (int k = 0; k < 2; ++k) {
      int  i = k ? i1 : i0;
      bool w = k ? w1 : w0;
      int src = i;
      if (w) {
        float ui = sh_u[i];
        int c = 0;
        for (int j = re_begin; j < end; ++j) {
          float uj = sh_u[j];
          c += (int)((uj < ui) | ((uj == ui) & (j < i)));
        }
        src = re_begin + c;
      }
      rs[i] = src;
    }
  }
}

// ---------------------------------------------------------------------------
// Kernel 2: bandwidth kernel. One wave per 512B row, one float4 per lane.
// mask_emb is staged into LDS via the CDNA5 async global->LDS path when the
// toolchain exposes the builtins (falls back to a plain copy otherwise).
// ---------------------------------------------------------------------------
#if __has_builtin(__builtin_amdgcn_global_load_async_to_lds_b128) && \
    __has_builtin(__builtin_amdgcn_s_wait_asynccnt)
#define USE_ASYNC_LDS 1
typedef int v4i __attribute__((ext_vector_type(4)));
typedef __attribute__((address_space(1))) v4i* gv4i_ptr;   // global
typedef __attribute__((address_space(3))) v4i* lv4i_ptr;   // LDS
#endif

__global__ __launch_bounds__(256)
void aug_copy_kernel(const float4* __restrict__ seq,
                     const float* __restrict__ mask_emb,
                     const int* __restrict__ rowsrc,
                     float4* __restrict__ out) {
  __shared__ float s_mask[ND];
  const int t = threadIdx.x;
  const int lane = t & 31;

  if (t < 32) {
#ifdef USE_ASYNC_LDS
    float* me = const_cast<float*>(mask_emb) + 4 * lane;
    __builtin_amdgcn_global_load_async_to_lds_b128(
        (gv4i_ptr)me, (lv4i_ptr)(&s_mask[4 * lane]), 0, 0);
    __builtin_amdgcn_s_wait_asynccnt(0);
#else
    ((float4*)s_mask)[lane] = ((const float4*)mask_emb)[lane];
#endif
  }
  __syncthreads();

  // Grid is sized exactly: NB*NL waves total, no remainder.
  size_t gwave = ((size_t)blockIdx.x * 256u + (size_t)t) >> 5;
  int src = rowsrc[gwave];                  // uniform per wave
  size_t brow = gwave >> 11;                // example index

  float4 v;
  if (src == -1) {
    v = make_float4(0.f, 0.f, 0.f, 0.f);
  } else if (src == -2) {
    v = ((const float4*)s_mask)[lane];
  } else {
    v = seq[(((brow << 11) + (size_t)src) << 5) + (size_t)lane];
  }
  out[(gwave << 5) + (size_t)lane] = v;
}

// ---------------------------------------------------------------------------
extern "C" void kernel_launch(void* const* d_in, const int* in_sizes, int n_in,
                              void* d_out, int out_size, void* d_ws,
                              size_t ws_size, hipStream_t stream) {
  const float* seq      = (const float*)d_in[0];   // [128,2048,128] f32
  const int*   seq_len  = (const int*)d_in[1];     // [128] i32
  const float* mask_emb = (const float*)d_in[2];   // [1,128] f32

  float* out     = (float*)d_out;
  float* out_len = out + (size_t)NB * NL * ND;     // aug_len, as float
  int*   rowsrc  = (int*)d_ws;                     // 128*2048*4 = 1 MB

  aug_plan_kernel<<<NB, 1024, 0, stream>>>(seq_len, rowsrc, out_len);
  aug_copy_kernel<<<(NB * NL) / 8, 256, 0, stream>>>(
      (const float4*)seq, mask_emb, rowsrc, (float4*)out);
}